// E61DecayGatedCell_4887672783105
// MI455X (gfx1250) — compile-verified
//
#include <hip/hip_runtime.h>
#include <cstddef>

// ---------------------------------------------------------------------------
// E61 DecayGatedCell for MI455X (gfx1250, wave32)
//
//   alpha = sigmoid(x @ Wa^T + ba)   [T,B,D]   (GEMM, fp32 WMMA 16x16x4)
//   v     =         x @ Wv^T + bv    [T,B,D]
//   h_t   = a_t * h_{t-1} + (1-a_t) * v_t      (chunked parallel scan)
//   out_t = h_t^2 * sigmoid(h_t)
//
// d_out = [ output (T*B*D) | h ((T+1)*B*D) ], fp32.
// alpha is staged in the output region, v in the h[1..T] slots; both are
// overwritten in-place by the scan (same thread reads then writes).
//
// GEMM data movement: double-buffered LDS with gfx1250 async global->LDS
// copies (GLOBAL_LOAD_ASYNC_TO_LDS_B128, tracked by ASYNCcnt).
// ---------------------------------------------------------------------------

typedef __attribute__((ext_vector_type(2))) float v2f;
typedef __attribute__((ext_vector_type(8))) float v8f;

#define T_STEPS 2048
#define BB      8
#define DD      1024
#define NBD     (BB * DD)        // 8192 lanes of the recurrence
#define MM      (T_STEPS * BB)   // 16384 GEMM rows

// GEMM tiling: block = 256 threads = 8 waves; tile = 128(M) x 64(N), K chunk 32.
#define BM 128
#define BN 64
#define BK 32
#define LDSS 36                  // padded LDS row stride (floats)

// Scan chunking
#define CHUNKS 32
#define CLEN   (T_STEPS / CHUNKS) // 64

#if defined(__has_builtin)
#if __has_builtin(__builtin_amdgcn_global_load_async_to_lds_b128)
#define HAVE_ASYNC_LDS 1
#endif
#endif
#ifndef HAVE_ASYNC_LDS
#define HAVE_ASYNC_LDS 0
#endif

#if HAVE_ASYNC_LDS
// Exact parameter type per clang diagnostic: GCC-style int vector of 4,
// global (AS1 / "__device__") for the source, LDS (AS3) for the destination.
typedef int v4i_gcc __attribute__((vector_size(16)));
typedef __attribute__((address_space(1))) v4i_gcc* gas_ptr;
typedef __attribute__((address_space(3))) v4i_gcc* las_ptr;
#define ASYNC_CP16(gp, lp) \
    __builtin_amdgcn_global_load_async_to_lds_b128((gas_ptr)(gp), (las_ptr)(lp), 0, 0)
#endif

__device__ __forceinline__ float sigf(float x) {
    return 1.0f / (1.0f + __expf(-x));
}

// ---------------------------------------------------------------------------
// Kernel 1: fused double GEMM, fp32 WMMA, double-buffered LDS pipeline.
// ---------------------------------------------------------------------------
__global__ __launch_bounds__(256)
void gemm_gates(const float* __restrict__ x,
                const float* __restrict__ Wa, const float* __restrict__ ba,
                const float* __restrict__ Wv, const float* __restrict__ bv,
                float* __restrict__ alpha_out,   // [MM, DD]
                float* __restrict__ v_out)       // [MM, DD] (== h region + NBD)
{
    __shared__ float As [2][BM * LDSS];
    __shared__ float Bsa[2][BN * LDSS];
    __shared__ float Bsv[2][BN * LDSS];

    const int tid  = threadIdx.x;
    const int wave = tid >> 5;        // wave32
    const int lane = tid & 31;
    const int lo   = lane & 15;
    const int hi   = lane >> 4;       // 0 or 1

    const int m0 = blockIdx.y * BM;
    const int n0 = blockIdx.x * BN;

    // Per-thread cooperative-copy coordinates (float4 granules).
    // A tile: 128x32 = 1024 granules -> 4/thread.  B tiles: 64x32 = 512 -> 2/thread.
    int rA[4], cA[4];
#pragma unroll
    for (int i = 0; i < 4; ++i) { int idx = tid + i * 256; rA[i] = idx >> 3; cA[i] = (idx & 7) * 4; }
    int rB[2], cB[2];
#pragma unroll
    for (int i = 0; i < 2; ++i) { int idx = tid + i * 256; rB[i] = idx >> 3; cB[i] = (idx & 7) * 4; }

    v8f zero = {};
    v8f acc_a[4], acc_v[4];
#pragma unroll
    for (int j = 0; j < 4; ++j) { acc_a[j] = zero; acc_v[j] = zero; }

#if HAVE_ASYNC_LDS
    // ---- async global->LDS pipeline (ASYNCcnt) ----
    // prologue: chunk 0 into buffer 0
#pragma unroll
    for (int i = 0; i < 4; ++i)
        ASYNC_CP16(x + (size_t)(m0 + rA[i]) * DD + 0 + cA[i],
                   &As[0][rA[i] * LDSS + cA[i]]);
#pragma unroll
    for (int i = 0; i < 2; ++i) {
        ASYNC_CP16(Wa + (size_t)(n0 + rB[i]) * DD + 0 + cB[i],
                   &Bsa[0][rB[i] * LDSS + cB[i]]);
        ASYNC_CP16(Wv + (size_t)(n0 + rB[i]) * DD + 0 + cB[i],
                   &Bsv[0][rB[i] * LDSS + cB[i]]);
    }
    asm volatile("s_wait_asynccnt 0x0" ::: "memory");
    __syncthreads();
#else
    // ---- fallback prologue: chunk 0 via VGPR staging ----
    {
#pragma unroll
        for (int i = 0; i < 4; ++i)
            *(float4*)(&As[0][rA[i] * LDSS + cA[i]]) =
                *(const float4*)(x + (size_t)(m0 + rA[i]) * DD + 0 + cA[i]);
#pragma unroll
        for (int i = 0; i < 2; ++i) {
            *(float4*)(&Bsa[0][rB[i] * LDSS + cB[i]]) =
                *(const float4*)(Wa + (size_t)(n0 + rB[i]) * DD + 0 + cB[i]);
            *(float4*)(&Bsv[0][rB[i] * LDSS + cB[i]]) =
                *(const float4*)(Wv + (size_t)(n0 + rB[i]) * DD + 0 + cB[i]);
        }
        __syncthreads();
    }
#endif

    for (int kc = 0; kc < DD; kc += BK) {
        const int cur = (kc / BK) & 1;
        const int nxt = cur ^ 1;
        const int kn  = kc + BK;
        const bool have_next = kn < DD;

#if HAVE_ASYNC_LDS
        if (have_next) {
#pragma unroll
            for (int i = 0; i < 4; ++i)
                ASYNC_CP16(x + (size_t)(m0 + rA[i]) * DD + kn + cA[i],
                           &As[nxt][rA[i] * LDSS + cA[i]]);
#pragma unroll
            for (int i = 0; i < 2; ++i) {
                ASYNC_CP16(Wa + (size_t)(n0 + rB[i]) * DD + kn + cB[i],
                           &Bsa[nxt][rB[i] * LDSS + cB[i]]);
                ASYNC_CP16(Wv + (size_t)(n0 + rB[i]) * DD + kn + cB[i],
                           &Bsv[nxt][rB[i] * LDSS + cB[i]]);
            }
        }
#else
        float4 regA[4], regBa[2], regBv[2];
        if (have_next) {
#pragma unroll
            for (int i = 0; i < 4; ++i)
                regA[i] = *(const float4*)(x + (size_t)(m0 + rA[i]) * DD + kn + cA[i]);
#pragma unroll
            for (int i = 0; i < 2; ++i) {
                regBa[i] = *(const float4*)(Wa + (size_t)(n0 + rB[i]) * DD + kn + cB[i]);
                regBv[i] = *(const float4*)(Wv + (size_t)(n0 + rB[i]) * DD + kn + cB[i]);
            }
        }
#endif

        // 8 K-steps of 4; per step: 1 A frag, 8 B frags, 8 WMMAs.
        // Fragment layout (ISA 7.12.2): lanes 0-15 hold K={k,k+1}, 16-31 K={k+2,k+3}.
        const float* as  = &As [cur][0];
        const float* bsa = &Bsa[cur][0];
        const float* bsv = &Bsv[cur][0];
#pragma unroll
        for (int kk = 0; kk < BK; kk += 4) {
            const int koff = kk + 2 * hi;
            v2f afrag = *(const v2f*)(&as[(wave * 16 + lo) * LDSS + koff]);
#pragma unroll
            for (int j = 0; j < 4; ++j) {
                v2f bfa = *(const v2f*)(&bsa[(j * 16 + lo) * LDSS + koff]);
                acc_a[j] = __builtin_amdgcn_wmma_f32_16x16x4_f32(
                    false, afrag, false, bfa, (short)0, acc_a[j], false, false);
                v2f bfv = *(const v2f*)(&bsv[(j * 16 + lo) * LDSS + koff]);
                acc_v[j] = __builtin_amdgcn_wmma_f32_16x16x4_f32(
                    false, afrag, false, bfv, (short)0, acc_v[j], false, false);
            }
        }

#if HAVE_ASYNC_LDS
        asm volatile("s_wait_asynccnt 0x0" ::: "memory");
        __syncthreads();
#else
        if (have_next) {
#pragma unroll
            for (int i = 0; i < 4; ++i)
                *(float4*)(&As[nxt][rA[i] * LDSS + cA[i]]) = regA[i];
#pragma unroll
            for (int i = 0; i < 2; ++i) {
                *(float4*)(&Bsa[nxt][rB[i] * LDSS + cB[i]]) = regBa[i];
                *(float4*)(&Bsv[nxt][rB[i] * LDSS + cB[i]]) = regBv[i];
            }
        }
        __syncthreads();
#endif
    }

    // Epilogue: D layout — VGPR r: lanes 0-15 -> M=r, lanes 16-31 -> M=r+8; N=lane&15.
#pragma unroll
    for (int j = 0; j < 4; ++j) {
        const int e = n0 + j * 16 + lo;
        const float bae = ba[e];
        const float bve = bv[e];
#pragma unroll
        for (int r = 0; r < 8; ++r) {
            const int m = m0 + wave * 16 + hi * 8 + r;
            alpha_out[(size_t)m * DD + e] = sigf(acc_a[j][r] + bae);
            v_out[(size_t)m * DD + e]     = acc_v[j][r] + bve;
        }
    }
}

// ---------------------------------------------------------------------------
// Kernel 2: per-chunk recurrence composition.  h_out = P*h_in + Q over CLEN steps.
// ---------------------------------------------------------------------------
__global__ __launch_bounds__(256)
void scan_phase1(const float* __restrict__ alpha,   // out region
                 const float* __restrict__ v,       // h region + NBD
                 float* __restrict__ Pws, float* __restrict__ Qws)
{
    const int g = blockIdx.x * 256 + threadIdx.x;   // 0..NBD/4-1
    const int c = blockIdx.y;
    const size_t base = (size_t)g * 4;

    float4 p = make_float4(1.f, 1.f, 1.f, 1.f);
    float4 q = make_float4(0.f, 0.f, 0.f, 0.f);
    const int t0 = c * CLEN;
#pragma unroll 4
    for (int t = t0; t < t0 + CLEN; ++t) {
        const float4 a = *(const float4*)(alpha + (size_t)t * NBD + base);
        const float4 w = *(const float4*)(v     + (size_t)t * NBD + base);
        q.x = a.x * q.x + (1.f - a.x) * w.x;  p.x *= a.x;
        q.y = a.y * q.y + (1.f - a.y) * w.y;  p.y *= a.y;
        q.z = a.z * q.z + (1.f - a.z) * w.z;  p.z *= a.z;
        q.w = a.w * q.w + (1.f - a.w) * w.w;  p.w *= a.w;
    }
    *(float4*)(Pws + (size_t)c * NBD + base) = p;
    *(float4*)(Qws + (size_t)c * NBD + base) = q;
}

// ---------------------------------------------------------------------------
// Kernel 3: serial combine over the 32 chunks -> chunk-start h; also emit h[0].
// ---------------------------------------------------------------------------
__global__ __launch_bounds__(256)
void scan_phase2(const float* __restrict__ h0,
                 const float* __restrict__ Pws, const float* __restrict__ Qws,
                 float* __restrict__ Hstart, float* __restrict__ h_out)
{
    const int g = blockIdx.x * 256 + threadIdx.x;
    const size_t base = (size_t)g * 4;

    float4 h = *(const float4*)(h0 + base);
    *(float4*)(h_out + base) = h;                    // h[0] = h0
#pragma unroll
    for (int c = 0; c < CHUNKS; ++c) {
        *(float4*)(Hstart + (size_t)c * NBD + base) = h;
        const float4 p = *(const float4*)(Pws + (size_t)c * NBD + base);
        const float4 q = *(const float4*)(Qws + (size_t)c * NBD + base);
        h.x = p.x * h.x + q.x;
        h.y = p.y * h.y + q.y;
        h.z = p.z * h.z + q.z;
        h.w = p.w * h.w + q.w;
    }
}

// ---------------------------------------------------------------------------
// Kernel 4: replay each chunk from its true start h; write h[t+1] (over v) and
// output[t] = h^2*sigmoid(h) (over alpha). Same thread reads then writes.
// ---------------------------------------------------------------------------
__global__ __launch_bounds__(256)
void scan_phase3(const float* __restrict__ Hstart,
                 float* __restrict__ out,     // alpha region -> output
                 float* __restrict__ hbuf)    // h region; v lives at slot t+1
{
    const int g = blockIdx.x * 256 + threadIdx.x;
    const int c = blockIdx.y;
    const size_t base = (size_t)g * 4;

    float4 h = *(const float4*)(Hstart + (size_t)c * NBD + base);
    const int t0 = c * CLEN;
#pragma unroll 2
    for (int t = t0; t < t0 + CLEN; ++t) {
        float* op = out  + (size_t)t * NBD + base;
        float* hp = hbuf + (size_t)(t + 1) * NBD + base;
        const float4 a = *(const float4*)op;
        const float4 w = *(const float4*)hp;
        h.x = a.x * h.x + (1.f - a.x) * w.x;
        h.y = a.y * h.y + (1.f - a.y) * w.y;
        h.z = a.z * h.z + (1.f - a.z) * w.z;
        h.w = a.w * h.w + (1.f - a.w) * w.w;
        *(float4*)hp = h;
        float4 o;
        o.x = h.x * h.x * sigf(h.x);
        o.y = h.y * h.y * sigf(h.y);
        o.z = h.z * h.z * sigf(h.z);
        o.w = h.w * h.w * sigf(h.w);
        *(float4*)op = o;
    }
}

// ---------------------------------------------------------------------------
extern "C" void kernel_launch(void* const* d_in, const int* in_sizes, int n_in,
                              void* d_out, int out_size, void* d_ws, size_t ws_size,
                              hipStream_t stream) {
    (void)in_sizes; (void)n_in; (void)out_size; (void)ws_size;

    const float* x  = (const float*)d_in[0];
    const float* h0 = (const float*)d_in[1];
    const float* Wa = (const float*)d_in[2];
    const float* ba = (const float*)d_in[3];
    const float* Wv = (const float*)d_in[4];
    const float* bv = (const float*)d_in[5];

    float* out    = (float*)d_out;                       // output [T*B*D]
    float* hreg   = out + (size_t)T_STEPS * NBD;         // h [(T+1)*B*D]
    float* vslots = hreg + NBD;                          // v staged at h[t+1]

    float* Pws    = (float*)d_ws;                        // CHUNKS*NBD floats (1 MB)
    float* Qws    = Pws + (size_t)CHUNKS * NBD;          // 1 MB
    float* Hstart = Qws + (size_t)CHUNKS * NBD;          // 1 MB

    // 1) GEMM: alpha -> out, v -> h[t+1] slots
    gemm_gates<<<dim3(DD / BN, MM / BM), 256, 0, stream>>>(
        x, Wa, ba, Wv, bv, out, vslots);

    // 2) per-chunk (P,Q)
    scan_phase1<<<dim3(NBD / 4 / 256, CHUNKS), 256, 0, stream>>>(
        out, vslots, Pws, Qws);

    // 3) serial chunk combine + h[0]
    scan_phase2<<<dim3(NBD / 4 / 256), 256, 0, stream>>>(
        h0, Pws, Qws, Hstart, hreg);

    // 4) replay chunks: h and output (in-place over v and alpha)
    scan_phase3<<<dim3(NBD / 4 / 256, CHUNKS), 256, 0, stream>>>(
        Hstart, out, hreg);
}